// AttentionLayer_16054587752892
// MI455X (gfx1250) — compile-verified
//
#include <hip/hip_runtime.h>
#include <hip/hip_bf16.h>
#include <cmath>
#include <cstdint>

#define N_B    8
#define C_DIM  64
#define L_DIM  512
#define D_DIM  64
#define HALF_W 32
#define TSTR   84   // LDS row stride (floats): 336B -> 16B aligned, conflict-free

typedef __attribute__((ext_vector_type(2))) float v2f;
typedef __attribute__((ext_vector_type(8))) float v8f;

// gfx1250 hardware tanh (TRANS32 pipe) -- one instruction vs ~20 for ocml tanhf
__device__ __forceinline__ float htanh(float v) {
  float r;
  asm("v_tanh_f32 %0, %1" : "=v"(r) : "v"(v));
  return r;
}

// ---------------------------------------------------------------------------
// Kernel 1: q = x^T @ Wt, k = x^T @ Wx   (per batch; x is (C,L) row-major)
// One wave computes one 16(M=l) x 16(N=d) tile; K=C=64 -> 16 wmma_f32_16x16x4.
// Both operands are lane-coalesced (lane index rides the contiguous dim), so
// direct global loads are the right path here.
// ---------------------------------------------------------------------------
__global__ __launch_bounds__(128) void qk_proj_kernel(
    const float* __restrict__ x, const float* __restrict__ Wx,
    const float* __restrict__ Wt, float* __restrict__ qout,
    float* __restrict__ kout)
{
  const int lane = threadIdx.x & 31;
  const int wid  = threadIdx.x >> 5;
  int t = blockIdx.x * 4 + wid;          // 0..2047
  const int sel = t >> 10;               // 0 -> q (Wt), 1 -> k (Wx)
  t &= 1023;
  const int n     = t >> 7;              // batch
  const int ltile = (t & 127) >> 2;      // 0..31
  const int dtile = t & 3;               // 0..3
  const int l0 = ltile * 16, d0 = dtile * 16;

  const float* W   = sel ? Wx : Wt;
  float*       out = sel ? kout : qout;

  const int half = lane >> 4;            // 0 or 1
  const int lm   = lane & 15;

  const float* xb = x + (size_t)n * C_DIM * L_DIM;

  v8f acc = {};
  for (int s = 0; s < 16; ++s) {
    const int ca = 4 * s + 2 * half;     // A K indices: ca, ca+1
    v2f a;
    a.x = xb[(size_t)ca * L_DIM + l0 + lm];
    a.y = xb[(size_t)(ca + 1) * L_DIM + l0 + lm];
    const int cb = 4 * s + half;         // B K indices: cb, cb+2
    v2f b;
    b.x = W[(size_t)cb * D_DIM + d0 + lm];
    b.y = W[(size_t)(cb + 2) * D_DIM + d0 + lm];
    acc = __builtin_amdgcn_wmma_f32_16x16x4_f32(false, a, false, b,
                                                (short)0, acc, false, false);
  }
  // D tile: vgpr r -> row M = r + 8*half, col = lm ; q/k layout (N,L,D)
  float* orow = out + ((size_t)n * L_DIM + l0) * D_DIM + d0 + lm;
#pragma unroll
  for (int r = 0; r < 8; ++r)
    orow[(size_t)(r + 8 * half) * D_DIM] = acc[r];
}

// ---------------------------------------------------------------------------
// Kernel 2: banded additive-attention scores + softmax.
// One 64-thread block per (n, i) row. Thread tid owns j = i-32+tid.
// Band-only max shift is mathematically identical to the reference's full-row
// max (the shift cancels in the normalization, up to the 1e-6 eps term).
// k rows read as float4 (b128, rows 256B aligned); tanh/exp on the hardware
// transcendental pipe (v_tanh_f32 / v_exp_f32).
// ---------------------------------------------------------------------------
__global__ __launch_bounds__(64) void score_kernel(
    const float* __restrict__ q, const float* __restrict__ kmat,
    const float* __restrict__ Wa, const float* __restrict__ bh,
    const float* __restrict__ ba, float* __restrict__ a_out)
{
  const int tid = threadIdx.x;           // 0..63
  const int row = blockIdx.x;            // 0..N*L-1
  const int n = row / L_DIM;
  const int i = row - n * L_DIM;

  __shared__ float qb[D_DIM];            // q row + bh, pre-summed
  __shared__ float was[D_DIM];
  __shared__ float red[64];
  __shared__ float band[64];

  qb[tid]  = q[((size_t)n * L_DIM + i) * D_DIM + tid] + bh[tid];
  was[tid] = Wa[tid];
  __syncthreads();

  const int  j     = i - HALF_W + tid;
  const bool valid = (j >= 0) && (j < L_DIM);
  const int  jc    = min(max(j, 0), L_DIM - 1);
  const float4* k4 = reinterpret_cast<const float4*>(
      kmat + ((size_t)n * L_DIM + jc) * D_DIM);

  float e = ba[0];
#pragma unroll
  for (int d4 = 0; d4 < D_DIM / 4; ++d4) {
    const float4 kv = k4[d4];
    const int d = d4 * 4;
    e = fmaf(was[d + 0], htanh(qb[d + 0] + kv.x), e);
    e = fmaf(was[d + 1], htanh(qb[d + 1] + kv.y), e);
    e = fmaf(was[d + 2], htanh(qb[d + 2] + kv.z), e);
    e = fmaf(was[d + 3], htanh(qb[d + 3] + kv.w), e);
  }
  if (!valid) e = -3.0e38f;

  // band max
  red[tid] = e;
  __syncthreads();
  for (int off = 32; off > 0; off >>= 1) {
    if (tid < off) red[tid] = fmaxf(red[tid], red[tid + off]);
    __syncthreads();
  }
  const float m = red[0];
  __syncthreads();

  const float p = valid ? __expf(e - m) : 0.0f;
  red[tid] = p;
  __syncthreads();
  for (int off = 32; off > 0; off >>= 1) {
    if (tid < off) red[tid] += red[tid + off];
    __syncthreads();
  }
  const float s = red[0];

  band[tid] = p / (s + 1e-6f);
  __syncthreads();

  float* arow = a_out + ((size_t)n * L_DIM + i) * L_DIM;
  for (int col = tid; col < L_DIM; col += 64) {
    const int off = col - (i - HALF_W);
    arow[col] = (off >= 0 && off < 64) ? band[off] : 0.0f;
  }
}

// ---------------------------------------------------------------------------
// Kernel 3: v[n,c,i] = sum_j a[n,i,j] * x[n,c,j]   (banded, j window = 80)
// One wave per 16(C) x 16(L_i) tile. Both operands index a different row per
// lane, so direct loads would scatter across 16 cache lines per instruction.
// Instead: coalesced async-copy (global_load_async_to_lds_b128, ASYNCcnt) of
// the two 16x80 tiles into LDS, then 20 K-steps of wmma_f32_16x16x4 fed from
// LDS. jbase is a multiple of 16, so every 16B chunk is fully in/out of
// range: out-of-range chunks are skipped (EXEC-masked VMEM is legal) and the
// tile is pre-zeroed. EXEC reconverges before the WMMAs.
// ---------------------------------------------------------------------------
__global__ __launch_bounds__(128) void av_kernel(
    const float* __restrict__ a, const float* __restrict__ x,
    float* __restrict__ vout)
{
  __shared__ float smem[4][2][16 * TSTR];   // [wave][x|a][tile], 43008 B

  const int lane = threadIdx.x & 31;
  const int wid  = threadIdx.x >> 5;
  int t = blockIdx.x * 4 + wid;          // 0..1023
  const int n  = t >> 7;
  const int ct = (t & 127) >> 5;         // 0..3
  const int it = t & 31;                 // 0..31
  const int c0 = ct * 16, i0 = it * 16;
  const int half = lane >> 4, lm = lane & 15;
  const int jbase = i0 - HALF_W;

  float* xls = &smem[wid][0][0];
  float* als = &smem[wid][1][0];

  // pre-zero this wave's tiles (covers band-edge chunks we skip below)
  for (int idx = lane; idx < 16 * TSTR; idx += 32) {
    xls[idx] = 0.0f;
    als[idx] = 0.0f;
  }
  asm volatile("s_wait_dscnt 0x0" ::: "memory");

  // coalesced async fill: 16 rows x 20 b128-chunks per tile, 32 lanes
  const float* xrow0 = x + (size_t)n * C_DIM * L_DIM + (size_t)c0 * L_DIM;
  const float* arow0 = a + ((size_t)n * L_DIM + i0) * L_DIM;
  for (int itr = 0; itr < 10; ++itr) {
    const int idx = itr * 32 + lane;     // 0..319
    const int row = idx / 20;
    const int ch  = idx - row * 20;
    const int j0  = jbase + ch * 4;
    const bool ok = (j0 >= 0) && (j0 < L_DIM);
    const unsigned ldsx =
        (unsigned)(uintptr_t)(xls + row * TSTR + ch * 4);
    const unsigned ldsa =
        (unsigned)(uintptr_t)(als + row * TSTR + ch * 4);
    const unsigned long long gx =
        (unsigned long long)(uintptr_t)(xrow0 + (size_t)row * L_DIM + j0);
    const unsigned long long ga =
        (unsigned long long)(uintptr_t)(arow0 + (size_t)row * L_DIM + j0);
    if (ok) {
      asm volatile("global_load_async_to_lds_b128 %0, %1, off"
                   :: "v"(ldsx), "v"(gx) : "memory");
      asm volatile("global_load_async_to_lds_b128 %0, %1, off"
                   :: "v"(ldsa), "v"(ga) : "memory");
    }
  }
  asm volatile("s_wait_asynccnt 0x0" ::: "memory");

  // WMMA from LDS: A[m=c][k=j], B[k=j][n=i]
  v8f acc = {};
  for (int s = 0; s < 20; ++s) {
    const int ca = 4 * s + 2 * half;     // local cols for A: ca, ca+1
    const int cb = 4 * s + half;         // local cols for B: cb, cb+2
    v2f av_, bv;
    av_.x = xls[lm * TSTR + ca];
    av_.y = xls[lm * TSTR + ca + 1];
    bv.x  = als[lm * TSTR + cb];
    bv.y  = als[lm * TSTR + cb + 2];
    acc = __builtin_amdgcn_wmma_f32_16x16x4_f32(false, av_, false, bv,
                                                (short)0, acc, false, false);
  }

  float* orow = vout + (size_t)n * C_DIM * L_DIM + (size_t)c0 * L_DIM + i0 + lm;
#pragma unroll
  for (int r = 0; r < 8; ++r)
    orow[(size_t)(r + 8 * half) * L_DIM] = acc[r];
}

// ---------------------------------------------------------------------------
extern "C" void kernel_launch(void* const* d_in, const int* in_sizes, int n_in,
                              void* d_out, int out_size, void* d_ws,
                              size_t ws_size, hipStream_t stream)
{
  (void)in_sizes; (void)n_in; (void)out_size; (void)ws_size;

  const float* x  = (const float*)d_in[0];
  const float* Wx = (const float*)d_in[1];
  const float* Wt = (const float*)d_in[2];
  const float* bh = (const float*)d_in[3];
  const float* Wa = (const float*)d_in[4];
  const float* ba = (const float*)d_in[5];

  float* out   = (float*)d_out;
  float* v_out = out;                                        // N*C*L
  float* a_out = out + (size_t)N_B * C_DIM * L_DIM;          // N*L*L

  float* q = (float*)d_ws;                                   // N*L*D
  float* k = q + (size_t)N_B * L_DIM * D_DIM;                // N*L*D

  qk_proj_kernel<<<512, 128, 0, stream>>>(x, Wx, Wt, q, k);
  score_kernel<<<N_B * L_DIM, 64, 0, stream>>>(q, k, Wa, bh, ba, a_out);
  av_kernel<<<256, 128, 0, stream>>>(a_out, x, v_out);
}